// IcosphereGradient_89713276879516
// MI455X (gfx1250) — compile-verified
//
#include <hip/hip_runtime.h>

// IcosphereGradient for MI455X (gfx1250).
// Bandwidth-bound stencil (~222 MB moved, ~0.25 GFLOP -> ~9.5us floor at
// 23.3 TB/s). Optimize data movement only:
//  - Tensor Data Mover: ONE tensor_load_to_lds per block stages 5 faces x
//    3 rows x 1024 fp32 (60 KB) into LDS via an iterating descriptor
//    (TENSORcnt path, CDNA5-specific). Fallback: per-lane async b128 loads.
//  - weight reuse across the 5 faces from registers (saves ~50 MB of traffic)
//  - non-temporal b128 stores for the streamed 168 MB output (keep L2 for x/w)

typedef float    v4f  __attribute__((ext_vector_type(4)));
typedef unsigned u32x4 __attribute__((ext_vector_type(4)));
typedef unsigned u32x8 __attribute__((ext_vector_type(8)));

#define NBATCH 4
#define NFACE  5
#define HSD    512
#define SD     1024

#define USE_TDM 1   // 0 -> per-lane global_load_async_to_lds_b128 staging

// Generic pointers to LDS carry the LDS byte offset in their low 32 bits
// (ISA 10.2 aperture rules).
__device__ __forceinline__ void async_ld_b128(const float* g, const float* lds) {
  unsigned loff = (unsigned)(unsigned long long)lds;
  asm volatile("global_load_async_to_lds_b128 %0, %1, off"
               :: "v"(loff), "v"(g)
               : "memory");
}
__device__ __forceinline__ void wait_async0() {
  asm volatile("s_wait_asynccnt 0" ::: "memory");
}
__device__ __forceinline__ void wait_tensor0() {
#if defined(__has_builtin) && __has_builtin(__builtin_amdgcn_s_wait_tensorcnt)
  __builtin_amdgcn_s_wait_tensorcnt(0);
#else
  asm volatile("s_wait_tensorcnt 0x0" ::: "memory");
#endif
}

// ---------------------------------------------------------------------------
// Interior: rows 1..HS-2, cols 1..S-2. One block per (batch, row); the block
// stages x[b, f, r-1..r+1, :] for all 5 faces into LDS via the TDM, loads
// its 24 weights once, then loops the 5 faces computing 4 columns per lane.
// ---------------------------------------------------------------------------
__global__ __launch_bounds__(256)
void grad_interior(const float* __restrict__ x, const float* __restrict__ wi,
                   float* __restrict__ out) {
  __shared__ float tile[NFACE][3][SD];       // 61440 B of the 320 KB WGP LDS

  const int r  = 1 + blockIdx.x;             // 1..510
  const int b  = blockIdx.y;                 // 0..3
  const int t  = threadIdx.x;                // 0..255
  const int c0 = 4 * t;                      // this lane's 4 columns

  const float* xb = x + (size_t)b * NFACE * HSD * SD;

#if USE_TDM
  // Wave 0 issues one TDM descriptor: 1024x3 fp32 tile, iterated 5x with
  // global step = one face (HS*S elements) and LDS step = 3 rows (bytes).
  if (t < 32) {
    unsigned long long ga = (unsigned long long)(xb + (size_t)(r - 1) * SD);
    unsigned ldsa = (unsigned)(unsigned long long)&tile[0][0][0];
    u32x4 g0; u32x8 g1; u32x4 g2; u32x4 g3;
    g0[0] = 1u;                                   // count=1 valid descriptor
    g0[1] = ldsa;                                 // lds_addr (bytes)
    g0[2] = (unsigned)ga;                         // global_addr[31:0]
    g0[3] = ((unsigned)(ga >> 32) & 0x01FFFFFFu)  // global_addr[56:32]
            | (2u << 30);                         // type = 2 ("image")
    g1[0] = (2u << 16) | (1u << 19);              // data_size=4B, iterate_en
    g1[1] = 1024u << 16;                          // tensor_dim0 = 1024
    g1[2] = 0x8000u << 16;                        // tensor_dim1 = 32768 (no OOB)
    g1[3] = 1024u << 16;                          // tile_dim0 = 1024
    g1[4] = 3u;                                   // tile_dim1 = 3, tile_dim2 = 0
    g1[5] = 1024u;                                // tensor_dim0_stride = 1024
    g1[6] = 0u;
    g1[7] = 0u;                                   // tensor_dim1_stride unused
    g2[0] = 0u;                                   // tensor_dim2
    g2[1] = 3u * SD * 4u;                         // lds_addr_increment / face
    g2[2] = (unsigned)(HSD * SD);                 // global_addr_increment / face
    g2[3] = 4u << 16;                             // iterate_count = 4 -> 5 tiles
    g3[0] = 0u; g3[1] = 0u; g3[2] = 0u; g3[3] = 0u;
    asm volatile("tensor_load_to_lds %0, %1, %2, %3"
                 :: "s"(g0), "s"(g1), "s"(g2), "s"(g3)
                 : "memory");
  }
#else
  for (int f = 0; f < NFACE; ++f)
    for (int dr = 0; dr < 3; ++dr)
      async_ld_b128(xb + ((size_t)f * HSD + (r - 1 + dr)) * SD + c0,
                    &tile[f][dr][c0]);
#endif

  // Weights overlap the DMA; reused across all 5 faces.
  // w_inner[k][r-1][c-1], clamped at the (discarded) frame columns.
  float w[6][4];
  for (int k = 0; k < 6; ++k) {
    const float* wp = wi + ((size_t)k * (HSD - 2) + (r - 1)) * (SD - 2);
    for (int j = 0; j < 4; ++j) {
      int cw = c0 + j - 1;
      cw = cw < 0 ? 0 : (cw > SD - 3 ? SD - 3 : cw);
      w[k][j] = wp[cw];
    }
  }

#if USE_TDM
  if (t < 32) wait_tensor0();   // DMA writes to LDS landed (wave 0 tracked it)
#else
  wait_async0();
#endif
  __syncthreads();              // release everyone once staging is complete

  const size_t CST = (size_t)NFACE * HSD * SD;   // component stride in out
  const int cm1 = (c0 == 0) ? 0 : c0 - 1;
  const int cp4 = (c0 + 4 > SD - 1) ? SD - 1 : c0 + 4;

  for (int f = 0; f < NFACE; ++f) {
    const float* tm = &tile[f][0][0];
    const float* tc = &tile[f][1][0];
    const float* tp = &tile[f][2][0];
    v4f rm = *(const v4f*)(tm + c0);
    v4f rc = *(const v4f*)(tc + c0);
    v4f rp = *(const v4f*)(tp + c0);
    float am[5] = {tm[cm1], rm.x, rm.y, rm.z, rm.w};            // row r-1: c0-1..c0+3
    float ac[6] = {tc[cm1], rc.x, rc.y, rc.z, rc.w, tc[cp4]};   // row r  : c0-1..c0+4
    float ap[5] = {rp.x, rp.y, rp.z, rp.w, tp[cp4]};            // row r+1: c0..c0+4

    float comp[4][4];
#pragma unroll
    for (int j = 0; j < 4; ++j) {
      float ctr = ac[j + 1];
      comp[0][j] = (am[j]     - ctr) * w[0][j];                               // (r-1,c-1)
      comp[1][j] = (ap[j + 1] - ctr) * w[1][j];                               // (r+1,c+1)
      comp[2][j] = (am[j + 1] - ctr) * w[2][j] + (ac[j + 2] - ctr) * w[3][j]; // (r-1,c)+(r,c+1)
      comp[3][j] = (ac[j]     - ctr) * w[4][j] + (ap[j]     - ctr) * w[5][j]; // (r,c-1)+(r+1,c)
    }

    float* ob = out + ((((size_t)b * 4) * NFACE + f) * HSD + r) * SD + c0;
    if (t != 0 && t != 255) {
#pragma unroll
      for (int k = 0; k < 4; ++k) {
        v4f v = {comp[k][0], comp[k][1], comp[k][2], comp[k][3]};
        __builtin_nontemporal_store(v, (v4f*)(ob + k * CST));   // b128 th:NT
      }
    } else {        // frame columns 0 / 1023 belong to the boundary kernel
#pragma unroll
      for (int k = 0; k < 4; ++k)
        for (int j = 0; j < 4; ++j) {
          int c = c0 + j;
          if (c >= 1 && c <= SD - 2)
            __builtin_nontemporal_store(comp[k][j], ob + k * CST + j);
        }
    }
  }
}

// ---------------------------------------------------------------------------
// Boundary frame: 11 wrap cases across faces. One thread per (b, f, position);
// positions: [0,S)=row0, [S,2S)=row HS-1, then col 0 rows 1..HS-2, col S-1.
// ---------------------------------------------------------------------------
#define NPOS (2 * SD + 2 * (HSD - 2))

__global__ __launch_bounds__(256)
void grad_boundary(const float* __restrict__ x, const float* __restrict__ poles,
                   const float* __restrict__ we,  const float* __restrict__ wcn,
                   const float* __restrict__ wne, const float* __restrict__ wnne,
                   const float* __restrict__ wnw, const float* __restrict__ wnww,
                   const float* __restrict__ wwe, const float* __restrict__ wsw,
                   const float* __restrict__ wsse,const float* __restrict__ wse,
                   const float* __restrict__ wcs, float* __restrict__ out) {
  int gid = blockIdx.x * blockDim.x + threadIdx.x;
  if (gid >= NBATCH * NFACE * NPOS) return;
  const int pos = gid % NPOS;
  const int fb  = gid / NPOS;
  const int f   = fb % NFACE;
  const int b   = fb / NFACE;
  const int fp  = (f + 4) % NFACE;   // previous face
  const int fn  = (f + 1) % NFACE;   // next face

  auto X = [&](int ff, int rr, int cc) -> float {
    return x[(((size_t)b * NFACE + ff) * HSD + rr) * SD + cc];
  };

  float c0v, c1v, c2v, c3v;
  int r, c;

  if (pos < SD) {                       // ---- row 0 ----
    r = 0; c = pos;
    float ctr = X(f, 0, c);
    if (c == 0) {                                        // di (wcs, 5)
      c0v = (X(fp, HSD - 1, HSD - 1) - ctr) * wcs[0];
      c1v = (X(f, 1, 1) - ctr) * wcs[1];
      c2v = (X(fp, HSD - 1, HSD) - ctr) * wcs[2] + (X(f, 0, 1) - ctr) * wcs[3];
      c3v = (X(f, 1, 0) - ctr) * wcs[4];
    } else if (c <= HSD - 1) {                           // ei (we, 6x511)
      int t = c - 1;
      c0v = (X(fp, HSD - 1, HSD + t) - ctr) * we[0 * 511 + t];
      c1v = (X(f, 1, t + 2) - ctr) * we[1 * 511 + t];
      c2v = (X(fp, HSD - 1, HSD + 1 + t) - ctr) * we[2 * 511 + t] +
            (X(f, 0, t + 2) - ctr) * we[3 * 511 + t];
      c3v = (X(f, 0, t) - ctr) * we[4 * 511 + t] +
            (X(f, 1, t + 1) - ctr) * we[5 * 511 + t];
    } else if (c == HSD) {                               // ci (wcn, 5)
      c0v = (X(fp, HSD - 1, SD - 1) - ctr) * wcn[0];
      c1v = (X(f, 1, HSD + 1) - ctr) * wcn[1];
      c2v = (X(f, 0, HSD + 1) - ctr) * wcn[2];
      c3v = (X(f, 0, HSD - 1) - ctr) * wcn[3] + (X(f, 1, HSD) - ctr) * wcn[4];
    } else if (c <= SD - 2) {                            // ni (wne, 6x510) rev wraps
      int t = c - (HSD + 1);
      c0v = (X(fp, HSD - 2 - t, SD - 1) - ctr) * wne[0 * 510 + t] +
            (X(fp, HSD - 1 - t, SD - 1) - ctr) * wne[1 * 510 + t];
      c1v = (X(f, 1, HSD + 2 + t) - ctr) * wne[2 * 510 + t] +
            (X(f, 1, HSD + 1 + t) - ctr) * wne[3 * 510 + t];
      c2v = (X(f, 0, HSD + 2 + t) - ctr) * wne[4 * 510 + t];
      c3v = (X(f, 0, HSD + t) - ctr) * wne[5 * 510 + t];
    } else {                                             // qi (wnne, 6) + pole
      c0v = (X(fp, 0, SD - 1) - ctr) * wnne[0] + (X(fp, 1, SD - 1) - ctr) * wnne[1];
      c1v = (X(fn, 0, SD - 1) - ctr) * wnne[2] + (X(f, 1, SD - 1) - ctr) * wnne[3];
      c2v = (poles[b * 2 + 1] - ctr) * wnne[4];
      c3v = (X(f, 0, SD - 2) - ctr) * wnne[5];
    }
  } else if (pos < 2 * SD) {            // ---- row HS-1 ----
    r = HSD - 1; c = pos - SD;
    float ctr = X(f, HSD - 1, c);
    if (c == 0) {                                        // pi (wsse, 5) + pole
      c0v = (X(fp, HSD - 1, 0) - ctr) * wsse[0];
      c1v = (X(fn, HSD - 1, 0) - ctr) * wsse[1];
      c2v = (X(f, HSD - 2, 0) - ctr) * wsse[2] + (X(f, HSD - 1, 1) - ctr) * wsse[3];
      c3v = (poles[b * 2 + 0] - ctr) * wsse[4];
    } else if (c <= HSD - 1) {                           // si (wsw, 6x511) rev wraps
      int t = c - 1;
      c0v = (X(f, HSD - 2, t) - ctr) * wsw[0 * 511 + t];
      c1v = (X(fn, HSD - 2 - t, 0) - ctr) * wsw[1 * 511 + t];
      c2v = (X(f, HSD - 2, 1 + t) - ctr) * wsw[2 * 511 + t] +
            (X(f, HSD - 1, 2 + t) - ctr) * wsw[3 * 511 + t];
      c3v = (X(f, HSD - 1, t) - ctr) * wsw[4 * 511 + t] +
            (X(fn, HSD - 1 - t, 0) - ctr) * wsw[5 * 511 + t];
    } else if (c <= SD - 2) {                            // ui (wwe, 6x511)
      int t = c - HSD;
      c0v = (X(f, HSD - 2, HSD - 1 + t) - ctr) * wwe[0 * 511 + t];
      c1v = (X(fn, 0, 1 + t) - ctr) * wwe[1 * 511 + t];
      c2v = (X(f, HSD - 2, HSD + t) - ctr) * wwe[2 * 511 + t] +
            (X(f, HSD - 1, HSD + 1 + t) - ctr) * wwe[3 * 511 + t];
      c3v = (X(f, HSD - 1, HSD - 1 + t) - ctr) * wwe[4 * 511 + t] +
            (X(fn, 0, t) - ctr) * wwe[5 * 511 + t];
    } else {                                             // gi (wnww, 6)
      c0v = (X(f, HSD - 2, SD - 2) - ctr) * wnww[0];
      c1v = (X(fn, 0, HSD) - ctr) * wnww[1];
      c2v = (X(f, HSD - 2, SD - 1) - ctr) * wnww[2] + (X(fn, 0, HSD + 1) - ctr) * wnww[3];
      c3v = (X(f, HSD - 1, SD - 2) - ctr) * wnww[4] + (X(fn, 0, HSD - 1) - ctr) * wnww[5];
    }
  } else if (pos < 2 * SD + (HSD - 2)) { // ---- col 0, rows 1..HS-2 (ri) ----
    int t = pos - 2 * SD; r = t + 1; c = 0;
    float ctr = X(f, r, 0);
    c0v = (X(fp, HSD - 1, HSD - 1 - t) - ctr) * wse[0 * 510 + t] +
          (X(fp, HSD - 1, HSD - 2 - t) - ctr) * wse[1 * 510 + t];
    c1v = (X(f, r, 1) - ctr) * wse[2 * 510 + t] + (X(f, r + 1, 1) - ctr) * wse[3 * 510 + t];
    c2v = (X(f, r - 1, 0) - ctr) * wse[4 * 510 + t];
    c3v = (X(f, r + 1, 0) - ctr) * wse[5 * 510 + t];
  } else {                               // ---- col S-1, rows 1..HS-2 (vi) ----
    int t = pos - (2 * SD + (HSD - 2)); r = t + 1; c = SD - 1;
    float ctr = X(f, r, SD - 1);
    c0v = (X(f, r - 1, SD - 2) - ctr) * wnw[0 * 510 + t];
    c1v = (X(fn, 0, SD - 2 - t) - ctr) * wnw[1 * 510 + t];
    c2v = (X(f, r - 1, SD - 1) - ctr) * wnw[2 * 510 + t] +
          (X(fn, 0, SD - 1 - t) - ctr) * wnw[3 * 510 + t];
    c3v = (X(f, r, SD - 2) - ctr) * wnw[4 * 510 + t] +
          (X(f, r + 1, SD - 1) - ctr) * wnw[5 * 510 + t];
  }

  const size_t CST = (size_t)NFACE * HSD * SD;
  size_t o = ((((size_t)b * 4) * NFACE + f) * HSD + r) * SD + c;
  out[o]           = c0v;
  out[o + CST]     = c1v;
  out[o + 2 * CST] = c2v;
  out[o + 3 * CST] = c3v;
}

// ---------------------------------------------------------------------------
extern "C" void kernel_launch(void* const* d_in, const int* in_sizes, int n_in,
                              void* d_out, int out_size, void* d_ws, size_t ws_size,
                              hipStream_t stream) {
  (void)in_sizes; (void)n_in; (void)out_size; (void)d_ws; (void)ws_size;
  const float* x     = (const float*)d_in[0];
  const float* poles = (const float*)d_in[1];
  const float* wi    = (const float*)d_in[2];
  const float* we    = (const float*)d_in[3];
  const float* wcn   = (const float*)d_in[4];
  const float* wne   = (const float*)d_in[5];
  const float* wnne  = (const float*)d_in[6];
  const float* wnw   = (const float*)d_in[7];
  const float* wnww  = (const float*)d_in[8];
  const float* wwe   = (const float*)d_in[9];
  const float* wsw   = (const float*)d_in[10];
  const float* wsse  = (const float*)d_in[11];
  const float* wse   = (const float*)d_in[12];
  const float* wcs   = (const float*)d_in[13];
  float* out = (float*)d_out;

  dim3 gridI(HSD - 2, NBATCH);                    // 510 x 4 blocks
  grad_interior<<<gridI, 256, 0, stream>>>(x, wi, out);

  int nb = NBATCH * NFACE * NPOS;                 // 61360 boundary elements
  grad_boundary<<<(nb + 255) / 256, 256, 0, stream>>>(
      x, poles, we, wcn, wne, wnne, wnw, wnww, wwe, wsw, wsse, wse, wcs, out);
}